// EncoderStage_22617297781336
// MI455X (gfx1250) — compile-verified
//
#include <hip/hip_runtime.h>
#include <hip/hip_bf16.h>

typedef __attribute__((ext_vector_type(2))) float v2f;
typedef __attribute__((ext_vector_type(8))) float v8f;

// ---------------------------------------------------------------------------
// One gather-GEMM-scatter pass for a single kernel offset.
//   out[scatter[r], :] += X[gather[r], :] @ W          (W is CIN x 64)
// One wave (32 lanes) computes a 16-row x 64-col tile with
// v_wmma_f32_16x16x4_f32 (exact fp32, matches reference math).
// W is staged in LDS pre-swizzled so each WMMA B fragment {W[2k,n],W[2k+1,n]}
// is a single aligned ds_load_b64 (no VGPR repacking).
// Scatter rows are unique within one offset, so plain RMW is race-free and
// deterministic; the "trash" row (index n_out) is absorbed by a pad row.
// ---------------------------------------------------------------------------
template <int CIN>
__global__ void spconv_pass(const float* __restrict__ X,      // [n_src(+1), CIN]
                            const float* __restrict__ W,      // [CIN, 64]
                            const int*  __restrict__ gather,  // [rows]
                            const int*  __restrict__ scatter, // [rows]
                            float* __restrict__ out,          // [n_out+1, 64]
                            int rows, int padg, int pads) {
  constexpr int KS = CIN / 4;                  // K-steps of 4 for f32 WMMA
  // Pair-swizzled weights: lw[((k>>1)*64 + n)*2 + (k&1)] = W[k*64 + n]
  __shared__ float lw[CIN * 64];
  for (int i = threadIdx.x; i < CIN * 64; i += blockDim.x) {
    int r = i >> 6, c = i & 63;
    lw[(((r >> 1) * 64 + c) << 1) | (r & 1)] = W[i];
  }
  __syncthreads();

  const int lane = threadIdx.x & 31;
  const int wave = threadIdx.x >> 5;
  const int tile = blockIdx.x * (blockDim.x >> 5) + wave;
  const int m    = lane & 15;                  // row within tile / col within block
  const int hi   = lane >> 4;                  // half-wave selector (K pair)
  const int r    = tile * 16 + m;

  int gidx = padg, sidx = pads;
  if (r < rows) { gidx = gather[r]; sidx = scatter[r]; }

  // Whole tile is padding (trailing pad region) -> nothing to do.
  if (__ballot(gidx != padg) == 0ull) return;

  // ---- A fragments: lane holds row (lane%16), K = {4s+2*hi, 4s+2*hi+1} ----
  v2f a[KS];
  if (gidx != padg) {
    const float* xp = X + (size_t)gidx * CIN + 2 * hi;
#pragma unroll
    for (int s = 0; s < KS; ++s) a[s] = *(const v2f*)(xp + 4 * s);
  } else {
#pragma unroll
    for (int s = 0; s < KS; ++s) { a[s].x = 0.0f; a[s].y = 0.0f; }
  }

  v8f acc[4] = {};                             // 4 col-blocks of 16

#pragma unroll
  for (int j = 0; j < 4; ++j) {
    // K-pair index for step s is (2s + hi); column n = 16j + m.
    const v2f* bp = (const v2f*)&lw[((hi * 64) + 16 * j + m) << 1];
#pragma unroll
    for (int s = 0; s < KS; ++s) {
      v2f b = bp[s * 128];                     // {W[4s+2hi, n], W[4s+2hi+1, n]}
      acc[j] = __builtin_amdgcn_wmma_f32_16x16x4_f32(
          false, a[s], false, b, (short)0, acc[j], false, false);
    }
  }

  // ---- scatter RMW: VGPR v holds row M = v + 8*hi, cols 16j..16j+15 ------
#pragma unroll
  for (int v = 0; v < 8; ++v) {
    const int row  = v + 8 * hi;
    const int srow = __shfl(sidx, row);        // lane 'row' holds its scatter idx
    float* p = out + (size_t)srow * 64 + m;
#pragma unroll
    for (int j = 0; j < 4; ++j) p[16 * j] += acc[j][v];
  }
}

// ---------------------------------------------------------------------------
__global__ void bn_relu_kernel(const float* __restrict__ in, float* __restrict__ o,
                               const float* __restrict__ g, const float* __restrict__ b,
                               const float* __restrict__ mean, const float* __restrict__ var,
                               int n_rows) {
  int i = blockIdx.x * blockDim.x + threadIdx.x;
  if (i >= n_rows * 64) return;
  int c = i & 63;
  float inv = g[c] * rsqrtf(var[c] + 1e-5f);
  float v = in[i] * inv + (b[c] - mean[c] * inv);
  o[i] = v > 0.0f ? v : 0.0f;
}

__global__ void zero_kernel(float* p, size_t n) {
  size_t i = (size_t)blockIdx.x * blockDim.x + threadIdx.x;
  size_t stride = (size_t)gridDim.x * blockDim.x;
  for (; i < n; i += stride) p[i] = 0.0f;
}

// ---------------------------------------------------------------------------
extern "C" void kernel_launch(void* const* d_in, const int* in_sizes, int n_in_args,
                              void* d_out, int out_size, void* d_ws, size_t ws_size,
                              hipStream_t stream) {
  const float* feat    = (const float*)d_in[0];
  const float* w_down  = (const float*)d_in[1];
  const float* gamma_d = (const float*)d_in[2];
  const float* beta_d  = (const float*)d_in[3];
  const float* mean_d  = (const float*)d_in[4];
  const float* var_d   = (const float*)d_in[5];
  const float* w_ref   = (const float*)d_in[6];
  const float* gamma_r = (const float*)d_in[7];
  const float* beta_r  = (const float*)d_in[8];
  const float* mean_r  = (const float*)d_in[9];
  const float* var_r   = (const float*)d_in[10];
  const int* gather_d  = (const int*)d_in[11];
  const int* scatter_d = (const int*)d_in[12];
  const int* gather_r  = (const int*)d_in[13];
  const int* scatter_r = (const int*)d_in[14];

  const int n_in   = in_sizes[0] / 32;   // feat: [n_in, 32]
  const int md     = in_sizes[11] / 16;  // gather_d: [16, md]
  const int mr     = in_sizes[13] / 81;  // gather_r: [81, mr]
  const int n_down = mr;                 // center offset hits every output row

  float* bufA = (float*)d_ws;                        // accumulator, [n_down+1, 64]
  float* y1   = bufA + (size_t)(n_down + 1) * 64;    // stage-1 activations, [n_down+1, 64]
  const size_t nElem = (size_t)(n_down + 1) * 64;

  // Zero accumulator + y1 (pad row of y1 must be zero for stage-2 gathers).
  zero_kernel<<<4096, 256, 0, stream>>>(bufA, 2 * nElem);

  // ---- Stage 1: downsample conv (16 parity offsets), C_in = 32 -----------
  {
    const int tiles  = (md + 15) / 16;
    const int blocks = (tiles + 7) / 8;
    for (int k = 0; k < 16; ++k) {
      spconv_pass<32><<<blocks, 256, 0, stream>>>(
          feat, w_down + (size_t)k * 32 * 64,
          gather_d + (size_t)k * md, scatter_d + (size_t)k * md,
          bufA, md, /*padg=*/n_in, /*pads=*/n_down);
    }
  }
  bn_relu_kernel<<<(n_down * 64 + 255) / 256, 256, 0, stream>>>(
      bufA, y1, gamma_d, beta_d, mean_d, var_d, n_down);

  // Reuse bufA as stage-2 accumulator.
  zero_kernel<<<4096, 256, 0, stream>>>(bufA, nElem);

  // ---- Stage 2: refine conv (81 offsets), C_in = 64 ----------------------
  {
    const int tiles  = (mr + 15) / 16;
    const int blocks = (tiles + 7) / 8;
    for (int k = 0; k < 81; ++k) {
      spconv_pass<64><<<blocks, 256, 0, stream>>>(
          y1, w_ref + (size_t)k * 64 * 64,
          gather_r + (size_t)k * mr, scatter_r + (size_t)k * mr,
          bufA, mr, /*padg=*/n_down, /*pads=*/n_down);
    }
  }
  bn_relu_kernel<<<(n_down * 64 + 255) / 256, 256, 0, stream>>>(
      bufA, (float*)d_out, gamma_r, beta_r, mean_r, var_r, n_down);
}